// ReconChessNet_38757784879215
// MI455X (gfx1250) — compile-verified
//
#include <hip/hip_runtime.h>
#include <math.h>

// ---------------------------------------------------------------------------
// CDNA5 (gfx1250) implementation of ReconChessNet PPO loss.
// All large GEMMs use V_WMMA_F32_16X16X4_F32 (wave32).
// Workspace requirement: ~497 MB of f32 scratch (dominated by x@Wk buffer).
// ---------------------------------------------------------------------------

typedef float v2f __attribute__((ext_vector_type(2)));
typedef float v8f __attribute__((ext_vector_type(8)));

#define LEAK 0.2f
#define CLIPC 0.2f

__device__ __forceinline__ v8f wmma4(v2f a, v2f b, v8f c) {
  // D = A(16x4 f32) * B(4x16 f32) + C(16x16 f32)
  return __builtin_amdgcn_wmma_f32_16x16x4_f32(false, a, false, b, (short)0, c,
                                               false, false);
}
__device__ __forceinline__ float leaky(float y) { return (y >= 0.f) ? y : LEAK * y; }
__device__ __forceinline__ float sigm(float x) { return 1.f / (1.f + __expf(-x)); }

// ---------------------------------------------------------------------------
// Kernel 1: step_mask + conv1(2x2,s2, 8->16) + conv2(2x2,s1, 16->32), fused.
// One wave per sample; 8 samples per 256-thread block.
// in: (65536, 13,8,8) NHWC(C=8) -> x2: (65536, 5,3,32)
// ---------------------------------------------------------------------------
__global__ __launch_bounds__(256) void conv12_mask_kernel(
    const float* __restrict__ inp, const float* __restrict__ w1,
    const float* __restrict__ b1, const float* __restrict__ w2,
    const float* __restrict__ b2, float* __restrict__ X2,
    float* __restrict__ maskBT) {
  __shared__ float sw1[512];
  __shared__ float sw2[2048];
  __shared__ float sb1[16];
  __shared__ float sb2[32];
  __shared__ float t1[8][384];  // conv1 output (6,4,16) per wave
  int tid = threadIdx.x;
  for (int i = tid; i < 512; i += 256) sw1[i] = w1[i];
  for (int i = tid; i < 2048; i += 256) sw2[i] = w2[i];
  if (tid < 16) sb1[tid] = b1[tid];
  if (tid < 32) sb2[tid] = b2[tid];
  __syncthreads();

  int wave = tid >> 5, lane = tid & 31;
  size_t s = (size_t)blockIdx.x * 8 + wave;
  const float* xin = inp + s * 832;

  // step_mask: any(input != 0)
  float nz = 0.f;
  for (int i = lane; i < 832; i += 32) nz += (xin[i] != 0.f) ? 1.f : 0.f;
  for (int off = 16; off > 0; off >>= 1) nz += __shfl_down(nz, off, 32);
  if (lane == 0) maskBT[s] = (nz > 0.f) ? 1.f : 0.f;

  // conv1: out (6,4,16) = 384 values, 12 per lane
  for (int i = 0; i < 12; ++i) {
    int o = lane + 32 * i;
    int oc = o & 15, ow = (o >> 4) & 3, oh = o >> 6;
    float acc = sb1[oc];
    for (int kh = 0; kh < 2; ++kh)
      for (int kw = 0; kw < 2; ++kw)
        for (int ic = 0; ic < 8; ++ic)
          acc += xin[((2 * oh + kh) * 8 + (2 * ow + kw)) * 8 + ic] *
                 sw1[((kh * 2 + kw) * 8 + ic) * 16 + oc];
    t1[wave][o] = leaky(acc);
  }
  __syncthreads();

  // conv2: out (5,3,32) = 480 values, 15 per lane
  const float* t = t1[wave];
  float* xo = X2 + s * 480;
  for (int i = 0; i < 15; ++i) {
    int o = lane + 32 * i;
    int oc = o & 31, ow = (o >> 5) % 3, oh = o / 96;
    float acc = sb2[oc];
    for (int kh = 0; kh < 2; ++kh)
      for (int kw = 0; kw < 2; ++kw)
        for (int ic = 0; ic < 16; ++ic)
          acc += t[((oh + kh) * 4 + (ow + kw)) * 16 + ic] *
                 sw2[((kh * 2 + kw) * 16 + ic) * 32 + oc];
    xo[o] = leaky(acc);
  }
}

// ---------------------------------------------------------------------------
// Kernel 2: conv3 as implicit-im2col WMMA GEMM.
// M = 131072 rows r=(sample,oh), K = 128 (kh,kw,ic), N = 64. Leaky epilogue.
// ---------------------------------------------------------------------------
__global__ __launch_bounds__(256) void conv3_gemm_kernel(
    const float* __restrict__ X2, const float* __restrict__ W3,
    const float* __restrict__ b3, float* __restrict__ T3) {
  int lane = threadIdx.x & 31;
  int tile = blockIdx.x * (blockDim.x >> 5) + (threadIdx.x >> 5);
  int m0 = (tile >> 2) << 4;  // 4 N-tiles (N=64)
  int n0 = (tile & 3) << 4;
  int hi = lane >> 4, l = lane & 15;
  int r = m0 + l;
  int sidx = r >> 1, oh = r & 1;
  const float* xbase = X2 + (size_t)sidx * 480 + oh * 192;
  v8f acc = {};
  for (int k0 = 0; k0 < 128; k0 += 4) {
    int kk = k0 + 2 * hi;
    int kh = kk >> 6, rem = kk & 63;
    const float* ap = xbase + kh * 96 + rem;
    v2f a; a.x = ap[0]; a.y = ap[1];
    const float* wp = W3 + (size_t)kk * 64 + n0 + l;
    v2f b; b.x = wp[0]; b.y = wp[64];
    acc = wmma4(a, b, acc);
  }
  float bv = b3[n0 + l];
  for (int v = 0; v < 8; ++v) {
    int row = m0 + v + hi * 8;
    T3[(size_t)row * 64 + n0 + l] = leaky(acc[v] + bv);
  }
}

// ---------------------------------------------------------------------------
// Generic WMMA GEMM: C[M,N] = act(A[M,K] @ W[K,N] + bias). Row-major.
// One wave per 16x16 output tile. Dims must be multiples of 16 / K of 4.
// ---------------------------------------------------------------------------
__global__ __launch_bounds__(256) void gemm_kernel(
    const float* __restrict__ A, const float* __restrict__ W,
    const float* __restrict__ bias, float* __restrict__ C, int M, int N, int K,
    int act) {
  int lane = threadIdx.x & 31;
  int tile = blockIdx.x * (blockDim.x >> 5) + (threadIdx.x >> 5);
  int ntiles = N >> 4;
  int m0 = (tile / ntiles) << 4;
  int n0 = (tile % ntiles) << 4;
  int hi = lane >> 4, l = lane & 15;
  const float* arow = A + (size_t)(m0 + l) * K;
  v8f acc = {};
  for (int k0 = 0; k0 < K; k0 += 4) {
    v2f a; a.x = arow[k0 + 2 * hi]; a.y = arow[k0 + 2 * hi + 1];
    const float* wp = W + (size_t)(k0 + 2 * hi) * N + n0 + l;
    v2f b; b.x = wp[0]; b.y = wp[N];
    acc = wmma4(a, b, acc);
  }
  float bv = bias ? bias[n0 + l] : 0.f;
  for (int v = 0; v < 8; ++v) {
    int row = m0 + v + hi * 8;
    float y = acc[v] + bv;
    if (act == 1) y = leaky(y);
    C[(size_t)row * N + n0 + l] = y;
  }
}

// ---------------------------------------------------------------------------
// Kernel: one LSTM time step. Each wave computes a 16(batch)x16(hidden) patch
// for ALL FOUR gates (4 WMMA accumulators) so the gate math fuses in-register.
// z = xWk[:,t,:] (bias pre-added) + h_in @ Wr.  512 tiles total.
// ---------------------------------------------------------------------------
__global__ __launch_bounds__(256) void lstm_step_kernel(
    const float* __restrict__ xWk, const float* __restrict__ Wr,
    const float* __restrict__ h_in, const float* __restrict__ c_in,
    const float* __restrict__ maskBT, float* __restrict__ h_out,
    float* __restrict__ c_out, int t) {
  int lane = threadIdx.x & 31;
  int tile = blockIdx.x * (blockDim.x >> 5) + (threadIdx.x >> 5);
  int m0 = (tile >> 4) << 4;  // batch tile (512/16 = 32)
  int j0 = (tile & 15) << 4;  // hidden tile (256/16 = 16)
  int hi = lane >> 4, l = lane & 15;
  const float* hrow = h_in + (size_t)(m0 + l) * 256;
  v8f ai = {}, af = {}, ag = {}, ao = {};
  for (int k0 = 0; k0 < 256; k0 += 4) {
    v2f a; a.x = hrow[k0 + 2 * hi]; a.y = hrow[k0 + 2 * hi + 1];
    const float* wp = Wr + (size_t)(k0 + 2 * hi) * 1024 + j0 + l;
    v2f b0, b1, b2, b3;
    b0.x = wp[0];   b0.y = wp[1024];
    b1.x = wp[256]; b1.y = wp[256 + 1024];
    b2.x = wp[512]; b2.y = wp[512 + 1024];
    b3.x = wp[768]; b3.y = wp[768 + 1024];
    ai = wmma4(a, b0, ai);
    af = wmma4(a, b1, af);
    ag = wmma4(a, b2, ag);
    ao = wmma4(a, b3, ao);
  }
  for (int v = 0; v < 8; ++v) {
    int b = m0 + v + hi * 8;
    int j = j0 + l;
    size_t xrow = ((size_t)b * 128 + t) * 1024;
    float zi = ai[v] + xWk[xrow + j];
    float zf = af[v] + xWk[xrow + 256 + j];
    float zg = ag[v] + xWk[xrow + 512 + j];
    float zo = ao[v] + xWk[xrow + 768 + j];
    size_t hc = (size_t)b * 256 + j;
    float c_old = c_in[hc];
    float h_old = h_in[hc];
    float c2 = sigm(zf) * c_old + sigm(zi) * tanhf(zg);
    float h2 = sigm(zo) * tanhf(c2);
    float m = maskBT[(size_t)b * 128 + t];
    h_out[hc] = (m != 0.f) ? h2 : h_old;
    c_out[hc] = (m != 0.f) ? c2 : c_old;
  }
}

// ---------------------------------------------------------------------------
// pir logits (N=36, not /16) + vpred, small VALU kernel. One block per batch.
// ---------------------------------------------------------------------------
__global__ __launch_bounds__(64) void pir_v_kernel(
    const float* __restrict__ h, const float* __restrict__ pir_w,
    const float* __restrict__ pir_b, const float* __restrict__ v_w,
    const float* __restrict__ v_b, float* __restrict__ pirL,
    float* __restrict__ vpred) {
  int b = blockIdx.x, j = threadIdx.x;
  const float* hr = h + (size_t)b * 256;
  if (j < 36) {
    float acc = pir_b[j];
    for (int k = 0; k < 256; ++k) acc += hr[k] * pir_w[k * 36 + j];
    pirL[b * 36 + j] = acc;
  } else if (j == 36) {
    float acc = v_b[0];
    for (int k = 0; k < 256; ++k) acc += hr[k] * v_w[k];
    vpred[b] = acc;
  }
}

// ---------------------------------------------------------------------------
// Row softmax -> masked q (pre global-normalization), plus per-row
// sum(q) and sum(q*log q) for the global-sum / entropy identity.
// One 256-thread block per row.
// ---------------------------------------------------------------------------
__global__ __launch_bounds__(256) void softmax_mask_kernel(
    const float* __restrict__ L, const unsigned char* __restrict__ mask,
    float* __restrict__ Q, float* __restrict__ sq, float* __restrict__ sql,
    int N) {
  __shared__ float red[256];
  int row = blockIdx.x, tid = threadIdx.x;
  const float* lr = L + (size_t)row * N;
  const unsigned char* mr = mask + (size_t)row * N;
  float mx = -INFINITY;
  for (int j = tid; j < N; j += 256) mx = fmaxf(mx, lr[j]);
  red[tid] = mx; __syncthreads();
  for (int s = 128; s > 0; s >>= 1) {
    if (tid < s) red[tid] = fmaxf(red[tid], red[tid + s]);
    __syncthreads();
  }
  mx = red[0]; __syncthreads();
  float z = 0.f;
  for (int j = tid; j < N; j += 256) z += __expf(lr[j] - mx);
  red[tid] = z; __syncthreads();
  for (int s = 128; s > 0; s >>= 1) {
    if (tid < s) red[tid] += red[tid + s];
    __syncthreads();
  }
  float Z = red[0]; __syncthreads();
  float sqa = 0.f, sqla = 0.f;
  for (int j = tid; j < N; j += 256) {
    float q = (__expf(lr[j] - mx) / Z) * (mr[j] ? 1.f : 0.f);
    Q[(size_t)row * N + j] = q;
    if (q > 0.f) { sqa += q; sqla += q * __logf(q); }
  }
  red[tid] = sqa; __syncthreads();
  for (int s = 128; s > 0; s >>= 1) {
    if (tid < s) red[tid] += red[tid + s];
    __syncthreads();
  }
  if (tid == 0) sq[row] = red[0];
  __syncthreads();
  red[tid] = sqla; __syncthreads();
  for (int s = 128; s > 0; s >>= 1) {
    if (tid < s) red[tid] += red[tid + s];
    __syncthreads();
  }
  if (tid == 0) sql[row] = red[0];
}

// ---------------------------------------------------------------------------
// Final single-block kernel: global sums, entropy identity, PPO pg/vf losses.
// Deterministic tree reductions (no float atomics).
// ---------------------------------------------------------------------------
__device__ float blk_sum512(float v, float* red) {
  int t = threadIdx.x;
  red[t] = v; __syncthreads();
  for (int s = 256; s > 0; s >>= 1) {
    if (t < s) red[t] += red[t + s];
    __syncthreads();
  }
  float r = red[0]; __syncthreads();
  return r;
}

__global__ __launch_bounds__(512) void finalize_kernel(
    const float* __restrict__ sqr, const float* __restrict__ sqlr,
    const float* __restrict__ sqm, const float* __restrict__ sqlm,
    const float* __restrict__ Qr, const float* __restrict__ Qm,
    const float* __restrict__ vpred, const int* __restrict__ a_taken,
    const float* __restrict__ lg_old, const float* __restrict__ GAE,
    const float* __restrict__ oldv, const float* __restrict__ rets,
    float* __restrict__ out) {
  __shared__ float red[512];
  int b = threadIdx.x;
  float S_r = blk_sum512(sqr[b], red);
  float E_r = blk_sum512(sqlr[b], red);
  float S_m = blk_sum512(sqm[b], red);
  float E_m = blk_sum512(sqlm[b], red);

  int a0 = a_taken[b * 4 + 0], a1 = a_taken[b * 4 + 1];
  int a2 = a_taken[b * 4 + 2], a3 = a_taken[b * 4 + 3];
  float pr = Qr[(size_t)b * 36 + a0 * 6 + a1] / S_r;
  float pm = Qm[(size_t)b * 4096 + ((a0 * 8 + a1) * 8 + a2) * 8 + a3] / S_m;
  float lg_new = ((b & 1) == 0) ? __logf(pr) : __logf(pm);
  float rt = __expf(lg_new - lg_old[b]);
  float g = GAE[b];
  float rtc = fminf(fmaxf(rt, 1.f - CLIPC), 1.f + CLIPC);
  float pg = fmaxf(-g * rt, -g * rtc);

  float vp = vpred[b], ov = oldv[b], rr = rets[b];
  float vc = ov + fminf(fmaxf(vp - ov, -CLIPC), CLIPC);
  float d1 = vp - rr, d2 = vc - rr;
  float vf = 0.5f * fmaxf(d1 * d1, d2 * d2);

  float pg_sum = blk_sum512(pg, red);
  float vf_sum = blk_sum512(vf, red);
  if (b == 0) {
    // sum(p log p) = (sum q log q)/S - log S,  p = q/S
    float entropy = -(E_r / S_r - __logf(S_r)) - (E_m / S_m - __logf(S_m));
    float pg_loss = pg_sum / 512.f;
    float vf_loss = vf_sum / 512.f;
    out[0] = pg_loss - entropy + vf_loss;
    out[1] = pg_loss;
    out[2] = entropy;
    out[3] = vf_loss;
  }
}

__global__ void zero_kernel(float* __restrict__ p, int n) {
  int i = blockIdx.x * blockDim.x + threadIdx.x;
  if (i < n) p[i] = 0.f;
}

// ---------------------------------------------------------------------------
extern "C" void kernel_launch(void* const* d_in, const int* in_sizes, int n_in,
                              void* d_out, int out_size, void* d_ws,
                              size_t ws_size, hipStream_t stream) {
  (void)in_sizes; (void)n_in; (void)out_size; (void)ws_size;
  const float* inputs   = (const float*)d_in[0];
  const unsigned char* mask_r = (const unsigned char*)d_in[1];
  const unsigned char* mask_m = (const unsigned char*)d_in[2];
  const float* lg_old   = (const float*)d_in[3];
  const int*   a_taken  = (const int*)d_in[4];
  const float* GAE      = (const float*)d_in[5];
  const float* oldv     = (const float*)d_in[6];
  const float* rets     = (const float*)d_in[7];
  const float* conv1_w  = (const float*)d_in[8];
  const float* conv1_b  = (const float*)d_in[9];
  const float* conv2_w  = (const float*)d_in[10];
  const float* conv2_b  = (const float*)d_in[11];
  const float* conv3_w  = (const float*)d_in[12];
  const float* conv3_b  = (const float*)d_in[13];
  const float* conv4_w  = (const float*)d_in[14];
  const float* conv4_b  = (const float*)d_in[15];
  const float* lstm_k   = (const float*)d_in[16];
  const float* lstm_rk  = (const float*)d_in[17];
  const float* lstm_b   = (const float*)d_in[18];
  const float* pir_w    = (const float*)d_in[19];
  const float* pir_b    = (const float*)d_in[20];
  const float* pim_w    = (const float*)d_in[21];
  const float* pim_b    = (const float*)d_in[22];
  const float* v_w      = (const float*)d_in[23];
  const float* v_b      = (const float*)d_in[24];
  float* out = (float*)d_out;
  float* ws = (float*)d_ws;

  // workspace layout (in floats)
  const size_t off_feat = 0;                       // 65536*256
  const size_t off_x2   = off_feat + 16777216ull;  // 65536*480
  const size_t off_t3   = off_x2 + 31457280ull;    // 131072*64
  const size_t off_xwk  = off_t3 + 8388608ull;     // 65536*1024
  const size_t off_mask = off_xwk + 67108864ull;   // 65536
  const size_t off_hA   = off_mask + 65536ull;     // 512*256
  const size_t off_cA   = off_hA + 131072ull;
  const size_t off_hB   = off_cA + 131072ull;
  const size_t off_cB   = off_hB + 131072ull;
  // heads alias the (dead by then) x2 region:
  const size_t off_pimL = off_x2;                  // 512*4096
  const size_t off_pirL = off_pimL + 2097152ull;   // 512*36
  const size_t off_vp   = off_pirL + 18432ull;     // 512
  const size_t off_Qr   = off_vp + 512ull;         // 512*36
  const size_t off_Qm   = off_Qr + 18432ull;       // 512*4096
  const size_t off_sqr  = off_Qm + 2097152ull;     // 512
  const size_t off_sqlr = off_sqr + 512ull;
  const size_t off_sqm  = off_sqlr + 512ull;
  const size_t off_sqlm = off_sqm + 512ull;

  float* feat = ws + off_feat;
  float* x2   = ws + off_x2;
  float* t3   = ws + off_t3;
  float* xwk  = ws + off_xwk;
  float* mbt  = ws + off_mask;

  // h0 = c0 = 0
  zero_kernel<<<(262144 + 255) / 256, 256, 0, stream>>>(ws + off_hA, 262144);

  // conv1+conv2 (+step_mask), fused VALU kernel: 65536 samples, 8/block
  conv12_mask_kernel<<<8192, 256, 0, stream>>>(inputs, conv1_w, conv1_b,
                                               conv2_w, conv2_b, x2, mbt);

  // conv3: M=131072, K=128, N=64 WMMA GEMM (implicit im2col), leaky
  conv3_gemm_kernel<<<4096, 256, 0, stream>>>(x2, conv3_w, conv3_b, t3);

  // conv4 (1x1): M=131072, K=64, N=128 -> feat (rows flatten to (65536,256))
  gemm_kernel<<<8192, 256, 0, stream>>>(t3, conv4_w, conv4_b, feat,
                                        131072, 128, 64, 1);

  // precompute x@Wk + b for all timesteps: M=65536, K=256, N=1024
  gemm_kernel<<<32768, 256, 0, stream>>>(feat, lstm_k, lstm_b, xwk,
                                         65536, 1024, 256, 0);

  // sequential LSTM scan, ping-pong h/c buffers
  float* h_cur = ws + off_hA; float* c_cur = ws + off_cA;
  float* h_nxt = ws + off_hB; float* c_nxt = ws + off_cB;
  for (int t = 0; t < 128; ++t) {
    lstm_step_kernel<<<64, 256, 0, stream>>>(xwk, lstm_rk, h_cur, c_cur, mbt,
                                             h_nxt, c_nxt, t);
    float* tp;
    tp = h_cur; h_cur = h_nxt; h_nxt = tp;
    tp = c_cur; c_cur = c_nxt; c_nxt = tp;
  }
  // h_cur = final hidden state (512 x 256)

  // heads
  gemm_kernel<<<1024, 256, 0, stream>>>(h_cur, pim_w, pim_b, ws + off_pimL,
                                        512, 4096, 256, 0);
  pir_v_kernel<<<512, 64, 0, stream>>>(h_cur, pir_w, pir_b, v_w, v_b,
                                       ws + off_pirL, ws + off_vp);

  // masked softmaxes + per-row partial sums
  softmax_mask_kernel<<<512, 256, 0, stream>>>(ws + off_pirL, mask_r,
                                               ws + off_Qr, ws + off_sqr,
                                               ws + off_sqlr, 36);
  softmax_mask_kernel<<<512, 256, 0, stream>>>(ws + off_pimL, mask_m,
                                               ws + off_Qm, ws + off_sqm,
                                               ws + off_sqlm, 4096);

  // final scalar losses
  finalize_kernel<<<1, 512, 0, stream>>>(
      ws + off_sqr, ws + off_sqlr, ws + off_sqm, ws + off_sqlm, ws + off_Qr,
      ws + off_Qm, ws + off_vp, a_taken, lg_old, GAE, oldv, rets, out);
}